// DiffusionGCN_52158082842768
// MI455X (gfx1250) — compile-verified
//
#include <hip/hip_runtime.h>

typedef __attribute__((ext_vector_type(2))) float v2f;
typedef __attribute__((ext_vector_type(8))) float v8f;

#define GCN_D 128   // feature width for layers (D == H == 128)

// ---------------------------------------------------------------------------
// fill: grid-stride constant fill (used for deg init and agg zeroing)
// ---------------------------------------------------------------------------
__global__ void gcn_fill_kernel(float* __restrict__ p, float v, long n) {
    long i = (long)blockIdx.x * blockDim.x + threadIdx.x;
    long stride = (long)gridDim.x * blockDim.x;
    for (; i < n; i += stride) p[i] = v;
}

// ---------------------------------------------------------------------------
// degree: deg[dst[e]] += 1  (deg pre-filled with 1.0 for the self-loop)
// ---------------------------------------------------------------------------
__global__ void gcn_degree_kernel(const int* __restrict__ dst,
                                  float* __restrict__ deg, int E) {
    int e = blockIdx.x * blockDim.x + threadIdx.x;
    if (e < E) unsafeAtomicAdd(&deg[dst[e]], 1.0f);
}

// dis[i] = rsqrt(deg[i])  (deg >= 1 always, self-loop included)
__global__ void gcn_rsqrt_kernel(float* __restrict__ deg, int n) {
    int i = blockIdx.x * blockDim.x + threadIdx.x;
    if (i < n) deg[i] = rsqrtf(deg[i]);
}

// ---------------------------------------------------------------------------
// Register-blocked GEMM via V_WMMA_F32_16X16X4_F32.
//   Out[rows x cols] = A[rows x K] @ W[cols x K]^T (+bias),  cols = NT*16
// K and NT are compile-time so all A/B loads fold their displacement into
// the 24-bit instruction IOFFSET (t*16*K*4 + k0*4 <= ~58 KB): no per-
// iteration 64-bit address arithmetic in the hot loop.
// One wave computes a full 16-row x cols band: NT accumulators, so each
// A float2 load feeds NT WMMAs (A read exactly once from memory; W is
// 64 KB and stays L2/WGP$-resident).
//   A 16x4 layout : lanes 0-15 -> M=lane, K={k0,k0+1}; lanes 16-31 -> K={k0+2,k0+3}
//   B 4x16 layout : lanes 0-15 -> N=lane, K={k0,k0+1}; lanes 16-31 -> K={k0+2,k0+3}
//   C/D 16x16     : VGPR r -> M=r (lanes 0-15) / M=r+8 (lanes 16-31), N=lane&15
// ---------------------------------------------------------------------------
template <int NT, int K>
__global__ void gcn_gemm_wmma_f32(const float* __restrict__ Ain,
                                  const float* __restrict__ W,
                                  float* __restrict__ Out,
                                  const float* __restrict__ bias,
                                  int rows) {
    constexpr int cols = NT * 16;
    int wave = (int)((blockIdx.x * blockDim.x + threadIdx.x) >> 5);
    int lane = threadIdx.x & 31;
    if (wave >= (rows >> 4)) return;         // uniform per wave: EXEC all-1s for WMMA

    int tileM = wave << 4;
    int l15   = lane & 15;
    int khalf = (lane >> 4) << 1;            // 0 or 2

    const float* aptr = Ain + (size_t)(tileM + l15) * K + khalf;
    const float* bptr = W   + (size_t)l15 * K + khalf;   // + t*16*K (imm) per tile

    v8f acc[NT];
#pragma unroll
    for (int t = 0; t < NT; ++t) acc[t] = (v8f){};

#pragma unroll 2
    for (int k0 = 0; k0 < K; k0 += 4) {
        v2f a = *(const v2f*)(aptr + k0);    // 8B aligned: row*K, khalf, k0 all even
#pragma unroll
        for (int t = 0; t < NT; ++t) {
            v2f b = *(const v2f*)(bptr + t * 16 * K + k0);  // constant displacement
            acc[t] = __builtin_amdgcn_wmma_f32_16x16x4_f32(
                         /*neg_a=*/false, a, /*neg_b=*/false, b,
                         /*c_mod=*/(short)0, acc[t],
                         /*reuse_a=*/false, /*reuse_b=*/false);
        }
    }

    int hi = lane >> 4;
    float* obase = Out + (size_t)(tileM + hi * 8) * cols + l15;
#pragma unroll
    for (int t = 0; t < NT; ++t) {
        float bb = bias ? bias[t * 16 + l15] : 0.0f;
#pragma unroll
        for (int r = 0; r < 8; ++r)
            obase[r * cols + t * 16] = acc[t][r] + bb;   // constant displacement
    }
}

// ---------------------------------------------------------------------------
// Edge scatter: one wave per edge; lane handles 4 consecutive features.
// Agg[dst] += M[src] * dis[src]*dis[dst]   via hardware global_atomic_add_f32
// ---------------------------------------------------------------------------
__global__ void gcn_scatter_kernel(const float* __restrict__ M,
                                   const int* __restrict__ src,
                                   const int* __restrict__ dst,
                                   const float* __restrict__ dis,
                                   float* __restrict__ Agg, int E) {
    long gid = (long)blockIdx.x * blockDim.x + threadIdx.x;
    int  e   = (int)(gid >> 5);
    if (e >= E) return;
    int lane = threadIdx.x & 31;

    int   s    = src[e];
    int   d    = dst[e];
    float norm = dis[s] * dis[d];

    float4 mv = ((const float4*)(M + (size_t)s * GCN_D))[lane];
    float* out = Agg + (size_t)d * GCN_D + lane * 4;
    unsafeAtomicAdd(out + 0, mv.x * norm);
    unsafeAtomicAdd(out + 1, mv.y * norm);
    unsafeAtomicAdd(out + 2, mv.z * norm);
    unsafeAtomicAdd(out + 3, mv.w * norm);
}

// ---------------------------------------------------------------------------
// Epilogue: add self-loop message (m * dis^2), bias, ReLU, residual.
// Safe when Out aliases Agg (pure per-element read-then-write).
// ---------------------------------------------------------------------------
__global__ void gcn_epilogue_kernel(const float* __restrict__ Agg,
                                    const float* __restrict__ M,
                                    const float* __restrict__ dis,
                                    const float* __restrict__ bias,
                                    const float* __restrict__ Hres,
                                    float* __restrict__ Out, long total) {
    long idx = (long)blockIdx.x * blockDim.x + threadIdx.x;
    if (idx >= total) return;
    int node = (int)(idx >> 7);              // /128
    int j    = (int)(idx & (GCN_D - 1));
    float sn = dis[node];
    sn *= sn;                                // self-loop norm = 1/deg
    float v = Agg[idx] + M[idx] * sn + bias[j];
    v = fmaxf(v, 0.0f);
    Out[idx] = v + Hres[idx];
}

// ---------------------------------------------------------------------------
extern "C" void kernel_launch(void* const* d_in, const int* in_sizes, int n_in,
                              void* d_out, int out_size, void* d_ws, size_t ws_size,
                              hipStream_t stream) {
    const float* x  = (const float*)d_in[0];
    const int*   ei = (const int*)d_in[1];
    const float* W1 = (const float*)d_in[2];
    const float* b1 = (const float*)d_in[3];
    const float* W2 = (const float*)d_in[4];
    const float* b2 = (const float*)d_in[5];
    const float* Wl = (const float*)d_in[6];
    const float* bl = (const float*)d_in[7];

    const int Dd = GCN_D;
    const int Nn = in_sizes[0] / Dd;         // 100000
    const int Ee = in_sizes[1] / 2;          // 1600000
    const int* srcIdx = ei;                  // edge_index[0,:]
    const int* dstIdx = ei + Ee;             // edge_index[1,:]

    float* ws      = (float*)d_ws;
    size_t nh      = (size_t)Nn * GCN_D;
    float* bufM    = ws;                     // linear-transform output m
    float* bufAgg  = bufM + nh;              // scatter accumulator (reused as h2)
    float* bufH1   = bufAgg + nh;            // layer-1 output
    float* disBuf  = bufH1 + nh;             // N floats: deg -> rsqrt(deg)

    const int  blk     = 256;
    const long totalNH = (long)nh;

    // ---- normalization coefficients ----
    gcn_fill_kernel<<<512, blk, 0, stream>>>(disBuf, 1.0f, (long)Nn);
    gcn_degree_kernel<<<(Ee + blk - 1) / blk, blk, 0, stream>>>(dstIdx, disBuf, Ee);
    gcn_rsqrt_kernel<<<(Nn + blk - 1) / blk, blk, 0, stream>>>(disBuf, Nn);

    const int bandWaves  = Nn / 16;                        // one wave per 16-row band
    const int gemmBlocks = (bandWaves + 7) / 8;            // 8 waves / 256-thread block
    const long scatThreads = (long)Ee * 32;
    const int scatBlocks = (int)((scatThreads + blk - 1) / blk);
    const int epiBlocks  = (int)((totalNH + blk - 1) / blk);

    // ---- layer 1 ----
    gcn_gemm_wmma_f32<8, GCN_D><<<gemmBlocks, blk, 0, stream>>>(x, W1, bufM, nullptr, Nn);
    gcn_fill_kernel<<<2048, blk, 0, stream>>>(bufAgg, 0.0f, totalNH);
    gcn_scatter_kernel<<<scatBlocks, blk, 0, stream>>>(bufM, srcIdx, dstIdx, disBuf, bufAgg, Ee);
    gcn_epilogue_kernel<<<epiBlocks, blk, 0, stream>>>(bufAgg, bufM, disBuf, b1, x, bufH1, totalNH);

    // ---- layer 2 ----
    gcn_gemm_wmma_f32<8, GCN_D><<<gemmBlocks, blk, 0, stream>>>(bufH1, W2, bufM, nullptr, Nn);
    gcn_fill_kernel<<<2048, blk, 0, stream>>>(bufAgg, 0.0f, totalNH);
    gcn_scatter_kernel<<<scatBlocks, blk, 0, stream>>>(bufM, srcIdx, dstIdx, disBuf, bufAgg, Ee);
    // h2 written in place over the accumulator
    gcn_epilogue_kernel<<<epiBlocks, blk, 0, stream>>>(bufAgg, bufM, disBuf, b2, bufH1, bufAgg, totalNH);

    // ---- final linear head (bias fused, 64 output classes -> NT=4) ----
    gcn_gemm_wmma_f32<4, GCN_D><<<gemmBlocks, blk, 0, stream>>>(bufAgg, Wl, (float*)d_out, bl, Nn);
}